// LIF_7069516169611
// MI455X (gfx1250) — compile-verified
//
#include <hip/hip_runtime.h>

typedef __attribute__((ext_vector_type(2))) float v2f;
typedef __attribute__((ext_vector_type(8))) float v8f;

// Problem constants (from reference setup_inputs)
constexpr int Bb   = 64;
constexpr int Tt   = 256;
constexpr int Dd   = 1024;   // K
constexpr int Nn   = 1024;   // N
constexpr int Mtot = Bb * Tt;  // 16384 GEMM rows

constexpr float THR   = 0.05f;
constexpr float DECAY = 0.01f;

// ---------------------------------------------------------------------------
// Kernel 1: i_in = x @ W   (fp32, exact, via V_WMMA_F32_16X16X4_F32)
// Each wave owns a 32(M) x 32(N) output tile: 4 accumulators of 16x16.
// ---------------------------------------------------------------------------
__global__ __launch_bounds__(256) void lif_gemm_f32(const float* __restrict__ X,
                                                    const float* __restrict__ W,
                                                    float* __restrict__ I) {
    const int gtid = blockIdx.x * blockDim.x + threadIdx.x;
    const int wave = gtid >> 5;          // global wave id
    const int lane = threadIdx.x & 31;

    constexpr int tilesN = Nn / 32;      // 32
    const int tm = wave / tilesN;        // 0..511
    const int tn = wave % tilesN;        // 0..31
    const int m0 = tm * 32;
    const int n0 = tn * 32;

    const int lmod = lane & 15;          // column / row within 16
    const int lhi  = lane >> 4;          // half-wave selector

    v8f acc00 = {}; v8f acc01 = {}; v8f acc10 = {}; v8f acc11 = {};

    // A fragment (16x4 fp32): lane<16 holds {K=k0,k0+1}, lane>=16 holds {K=k0+2,k0+3}
    // for row m0+lmod -> one contiguous float2 load per fragment.
    const float* A0 = X + (size_t)(m0 + lmod) * Dd + 2 * lhi;        // M-tile 0
    const float* A1 = A0 + (size_t)16 * Dd;                          // M-tile 1

    // B fragment (4x16 fp32): VGPR0 = row (k0 + 2*lhi), VGPR1 = row (k0 + 2*lhi + 1),
    // column n0+lmod -> two b32 loads strided by Nn.
    const float* B0 = W + (size_t)(2 * lhi) * Nn + n0 + lmod;        // N-tile 0
    const float* B1 = B0 + 16;                                       // N-tile 1

    for (int k0 = 0; k0 < Dd; k0 += 4) {
        v2f a0 = *(const v2f*)(A0 + k0);
        v2f a1 = *(const v2f*)(A1 + k0);

        const float* bp0 = B0 + (size_t)k0 * Nn;
        const float* bp1 = B1 + (size_t)k0 * Nn;
        v2f b0, b1;
        b0.x = bp0[0]; b0.y = bp0[Nn];
        b1.x = bp1[0]; b1.y = bp1[Nn];

        acc00 = __builtin_amdgcn_wmma_f32_16x16x4_f32(false, a0, false, b0,
                                                      (short)0, acc00, false, false);
        acc01 = __builtin_amdgcn_wmma_f32_16x16x4_f32(false, a0, false, b1,
                                                      (short)0, acc01, false, false);
        acc10 = __builtin_amdgcn_wmma_f32_16x16x4_f32(false, a1, false, b0,
                                                      (short)0, acc10, false, false);
        acc11 = __builtin_amdgcn_wmma_f32_16x16x4_f32(false, a1, false, b1,
                                                      (short)0, acc11, false, false);
    }

    // C/D layout: VGPR j, lanes 0-15 -> M=j, N=lane; lanes 16-31 -> M=j+8, N=lane-16
    {
        float* c = I + (size_t)(m0 + 8 * lhi) * Nn + (n0 + lmod);
        #pragma unroll
        for (int j = 0; j < 8; ++j) { c[(size_t)j * Nn] = acc00[j]; c[(size_t)j * Nn + 16] = acc01[j]; }
        c += (size_t)16 * Nn;
        #pragma unroll
        for (int j = 0; j < 8; ++j) { c[(size_t)j * Nn] = acc10[j]; c[(size_t)j * Nn + 16] = acc11[j]; }
    }
}

// ---------------------------------------------------------------------------
// Kernel 2: sequential LIF scan over T. One thread per (b, n); coalesced in n.
// v_t = d*v + (1-d)*i_t - THR*z ; z_t = (v_t > THR) ? 1 : 0
// ---------------------------------------------------------------------------
__global__ __launch_bounds__(256) void lif_scan(const float* __restrict__ I,
                                                float* __restrict__ Vout,
                                                float* __restrict__ Zout) {
    const int tid = blockIdx.x * blockDim.x + threadIdx.x;   // 0 .. B*N-1
    const int b = tid >> 10;       // / Nn
    const int n = tid & (Nn - 1);

    const size_t base = (size_t)b * Tt * Nn + n;
    const float* ip = I + base;
    float* vp = Vout + base;
    float* zp = Zout + base;

    float v = 0.0f, z = 0.0f;
    for (int t = 0; t < Tt; ++t) {
        const float it = ip[(size_t)t * Nn];
        v = DECAY * v + (1.0f - DECAY) * it - THR * z;
        z = (v > THR) ? 1.0f : 0.0f;
        vp[(size_t)t * Nn] = v;
        zp[(size_t)t * Nn] = z;
    }
}

// ---------------------------------------------------------------------------
extern "C" void kernel_launch(void* const* d_in, const int* in_sizes, int n_in,
                              void* d_out, int out_size, void* d_ws, size_t ws_size,
                              hipStream_t stream) {
    const float* x = (const float*)d_in[0];   // [B, T, D]
    const float* W = (const float*)d_in[1];   // [D, N]
    float* I = (float*)d_ws;                  // [B*T, N] intermediate (64 MB)
    float* outV = (float*)d_out;                          // [B, T, N]
    float* outZ = outV + (size_t)Bb * Tt * Nn;            // [B, T, N]

    // GEMM: (Mtot/32)*(Nn/32) waves = 16384 waves = 524288 threads
    const int gemmThreads = (Mtot / 32) * (Nn / 32) * 32;
    lif_gemm_f32<<<gemmThreads / 256, 256, 0, stream>>>(x, W, I);

    // Scan: B*N threads
    lif_scan<<<(Bb * Nn) / 256, 256, 0, stream>>>(I, outV, outZ);
}